// MambaBlock_78400333021354
// MI455X (gfx1250) — compile-verified
//
#include <hip/hip_runtime.h>
#include <math.h>

#define D_MODEL 1024
#define D_STATE 16
#define D_CONV  4
#define DT_RANK 64
#define D_INNER 2048
#define LSEQ    2048

typedef __attribute__((ext_vector_type(2))) float v2f;
typedef __attribute__((ext_vector_type(8))) float v8f;

// Fragment group for one K=4 step: 2 A-fragments (two M-tiles) + 4 B-fragments
struct Frag { v2f a0, a1, b0, b1, b2, b3; };

__device__ __forceinline__ void load_frag(
    Frag& f, const float* Ar0, const float* Ar1,
    const float* w0, const float* w1, const float* w2, const float* w3, int k)
{
    f.a0 = *(const v2f*)(Ar0 + k);
    f.a1 = *(const v2f*)(Ar1 + k);
    f.b0 = *(const v2f*)(w0 + k);
    f.b1 = *(const v2f*)(w1 + k);
    f.b2 = *(const v2f*)(w2 + k);
    f.b3 = *(const v2f*)(w3 + k);
}

__device__ __forceinline__ void do_wmma(const Frag& f, v8f acc[2][4])
{
    acc[0][0] = __builtin_amdgcn_wmma_f32_16x16x4_f32(false, f.a0, false, f.b0, (short)0, acc[0][0], false, false);
    acc[1][0] = __builtin_amdgcn_wmma_f32_16x16x4_f32(false, f.a1, false, f.b0, (short)0, acc[1][0], false, false);
    acc[0][1] = __builtin_amdgcn_wmma_f32_16x16x4_f32(false, f.a0, false, f.b1, (short)0, acc[0][1], false, false);
    acc[1][1] = __builtin_amdgcn_wmma_f32_16x16x4_f32(false, f.a1, false, f.b1, (short)0, acc[1][1], false, false);
    acc[0][2] = __builtin_amdgcn_wmma_f32_16x16x4_f32(false, f.a0, false, f.b2, (short)0, acc[0][2], false, false);
    acc[1][2] = __builtin_amdgcn_wmma_f32_16x16x4_f32(false, f.a1, false, f.b2, (short)0, acc[1][2], false, false);
    acc[0][3] = __builtin_amdgcn_wmma_f32_16x16x4_f32(false, f.a0, false, f.b3, (short)0, acc[0][3], false, false);
    acc[1][3] = __builtin_amdgcn_wmma_f32_16x16x4_f32(false, f.a1, false, f.b3, (short)0, acc[1][3], false, false);
}

// ---------------------------------------------------------------------------
// fp32 WMMA GEMM, register-blocked + double-buffered software pipeline:
//   C[M,N] = act( A[M,K(lda)] * W[N,K]^T (+bias) )
// One wave: 32(M) x 64(N) macro-tile = 2x4 accumulators, 8 wmmas / K=4 step.
// Steady state: loads for group k+4 issued before wmmas of group k execute.
// Requires K % 16 == 0 and K >= 16 (holds for all uses here).
// N-edge: B row index clamped to N-1; clamped tiles are never stored.
// act: 0 = none, 1 = softplus(v + bias[col])
// ---------------------------------------------------------------------------
__global__ __launch_bounds__(128, 1) void gemm_wmma_f32(
    const float* __restrict__ A, const float* __restrict__ W,
    const float* __restrict__ bias, float* __restrict__ C,
    int M, int N, int K, int lda, int act)
{
    const int lane = threadIdx.x & 31;
    const int wave = threadIdx.x >> 5;
    const int m0   = (blockIdx.y << 7) + (wave << 5);   // wave's 32-row strip
    const int n0   = blockIdx.x << 6;                   // 64-col strip
    if (m0 >= M) return;

    const int mrow  = lane & 15;
    const int khalf = (lane >> 4) << 1;                 // 0 | 2

    const float* __restrict__ Ar0 = A + (size_t)(m0 + mrow) * lda + khalf;
    const float* __restrict__ Ar1 = Ar0 + (size_t)16 * lda;

    const float* Wr[4];
#pragma unroll
    for (int j = 0; j < 4; ++j) {
        int row = n0 + (j << 4) + mrow;
        row = (row < N) ? row : (N - 1);
        Wr[j] = W + (size_t)row * K + khalf;
    }

    v8f acc[2][4];
#pragma unroll
    for (int i = 0; i < 2; ++i)
#pragma unroll
        for (int j = 0; j < 4; ++j) acc[i][j] = (v8f){};

    Frag f0, f1;
    load_frag(f0, Ar0, Ar1, Wr[0], Wr[1], Wr[2], Wr[3], 0);

    int k0 = 0;
    for (; k0 < K - 16; k0 += 16) {
        __builtin_prefetch(Ar0   + k0 + 128, 0, 1);   // global_prefetch_b8
        __builtin_prefetch(Ar1   + k0 + 128, 0, 1);
        __builtin_prefetch(Wr[0] + k0 + 128, 0, 1);
        __builtin_prefetch(Wr[2] + k0 + 128, 0, 1);
        load_frag(f1, Ar0, Ar1, Wr[0], Wr[1], Wr[2], Wr[3], k0 + 4);
        do_wmma(f0, acc);
        load_frag(f0, Ar0, Ar1, Wr[0], Wr[1], Wr[2], Wr[3], k0 + 8);
        do_wmma(f1, acc);
        load_frag(f1, Ar0, Ar1, Wr[0], Wr[1], Wr[2], Wr[3], k0 + 12);
        do_wmma(f0, acc);
        load_frag(f0, Ar0, Ar1, Wr[0], Wr[1], Wr[2], Wr[3], k0 + 16);
        do_wmma(f1, acc);
    }
    // peeled final 16-K chunk (no load beyond K)
    load_frag(f1, Ar0, Ar1, Wr[0], Wr[1], Wr[2], Wr[3], k0 + 4);
    do_wmma(f0, acc);
    load_frag(f0, Ar0, Ar1, Wr[0], Wr[1], Wr[2], Wr[3], k0 + 8);
    do_wmma(f1, acc);
    load_frag(f1, Ar0, Ar1, Wr[0], Wr[1], Wr[2], Wr[3], k0 + 12);
    do_wmma(f0, acc);
    do_wmma(f1, acc);

    const int rsub = (lane >> 4) << 3;                  // 0 | 8
#pragma unroll
    for (int j = 0; j < 4; ++j) {
        const int col = n0 + (j << 4) + mrow;
        if ((n0 + (j << 4)) >= N) continue;             // skip clamped tiles
        const float bv = (act == 1) ? bias[col] : 0.0f;
#pragma unroll
        for (int i = 0; i < 2; ++i) {
            const int rbase = m0 + (i << 4) + rsub;
#pragma unroll
            for (int r = 0; r < 8; ++r) {
                float v = acc[i][j][r];
                if (act == 1) {
                    v += bv;
                    v = (v > 20.0f) ? v : log1pf(__expf(v));   // softplus
                }
                C[(size_t)(rbase + r) * N + col] = v;
            }
        }
    }
}

// ---------------------------------------------------------------------------
// Depthwise causal conv1d (4 taps) + SiLU.  x_in = first half of xz rows.
// ---------------------------------------------------------------------------
__global__ __launch_bounds__(256) void conv_silu_kernel(
    const float* __restrict__ xz, const float* __restrict__ cw,
    const float* __restrict__ cb, float* __restrict__ xc)
{
    int idx = blockIdx.x * blockDim.x + threadIdx.x;     // l * D_INNER + d
    if (idx >= LSEQ * D_INNER) return;
    int l = idx >> 11;                                   // / D_INNER
    int d = idx & (D_INNER - 1);

    float acc = cb[d];
#pragma unroll
    for (int j = 0; j < D_CONV; ++j) {
        int ls = l - (D_CONV - 1) + j;
        if (ls >= 0)
            acc += cw[d * D_CONV + j] * xz[(size_t)ls * (2 * D_INNER) + d];
    }
    xc[idx] = acc / (1.0f + __expf(-acc));               // silu
}

// ---------------------------------------------------------------------------
// Selective-scan: one thread per d_inner channel, h[16] in registers.
// B/C rows (shared by all channels) staged through LDS, 32 steps per chunk.
// Fuses  y += x*D,  y *= silu(z),  and writes gated y in place over x_conv.
// ---------------------------------------------------------------------------
#define SCAN_TS 32
__global__ __launch_bounds__(256) void ssm_scan_kernel(
    const float* __restrict__ xdbl, const float* __restrict__ dtbuf,
    const float* __restrict__ xz,   const float* __restrict__ A_log,
    const float* __restrict__ D_param, float* __restrict__ xc_io)
{
    __shared__ float sB[SCAN_TS][D_STATE];
    __shared__ float sC[SCAN_TS][D_STATE];

    const int d = blockIdx.x * blockDim.x + threadIdx.x;   // 0..D_INNER-1

    float Acoef[D_STATE];
#pragma unroll
    for (int n = 0; n < D_STATE; ++n)
        Acoef[n] = -__expf(A_log[(size_t)d * D_STATE + n]);
    const float Dp = D_param[d];

    float h[D_STATE];
#pragma unroll
    for (int n = 0; n < D_STATE; ++n) h[n] = 0.0f;

    for (int lc = 0; lc < LSEQ; lc += SCAN_TS) {
        __syncthreads();   // previous chunk done reading sB/sC
        for (int i = threadIdx.x; i < SCAN_TS * D_STATE; i += blockDim.x) {
            int t = i >> 4, n = i & 15;
            const float* row = xdbl + (size_t)(lc + t) * (DT_RANK + 2 * D_STATE);
            sB[t][n] = row[DT_RANK + n];
            sC[t][n] = row[DT_RANK + D_STATE + n];
        }
        __syncthreads();

        for (int t = 0; t < SCAN_TS; ++t) {
            const int l = lc + t;
            const float dt = dtbuf[(size_t)l * D_INNER + d];
            const float xv = xc_io[(size_t)l * D_INNER + d];
            float y = 0.0f;
#pragma unroll
            for (int n = 0; n < D_STATE; ++n) {
                float dA = __expf(dt * Acoef[n]);
                h[n] = dA * h[n] + (dt * sB[t][n]) * xv;
                y += h[n] * sC[t][n];
            }
            y += xv * Dp;
            float z = xz[(size_t)l * (2 * D_INNER) + D_INNER + d];
            float g = z / (1.0f + __expf(-z));             // silu(z)
            xc_io[(size_t)l * D_INNER + d] = y * g;        // gated y, in place
        }
    }
}

// ---------------------------------------------------------------------------
extern "C" void kernel_launch(void* const* d_in, const int* in_sizes, int n_in,
                              void* d_out, int out_size, void* d_ws, size_t ws_size,
                              hipStream_t stream) {
    const float* x       = (const float*)d_in[0];  // (L, D_MODEL)
    const float* W_in    = (const float*)d_in[1];  // (2*D_INNER, D_MODEL)
    const float* conv_w  = (const float*)d_in[2];  // (D_INNER, 1, 4)
    const float* conv_b  = (const float*)d_in[3];  // (D_INNER,)
    const float* W_x     = (const float*)d_in[4];  // (96, D_INNER)
    const float* W_dt    = (const float*)d_in[5];  // (D_INNER, DT_RANK)
    const float* b_dt    = (const float*)d_in[6];  // (D_INNER,)
    const float* A_log   = (const float*)d_in[7];  // (D_INNER, D_STATE)
    const float* D_param = (const float*)d_in[8];  // (D_INNER,)
    const float* W_out   = (const float*)d_in[9];  // (D_MODEL, D_INNER)
    float* out = (float*)d_out;

    // workspace layout (floats)
    float* xz   = (float*)d_ws;                         // L x 2*D_INNER
    float* xc   = xz   + (size_t)LSEQ * 2 * D_INNER;    // L x D_INNER (-> gated y)
    float* xdbl = xc   + (size_t)LSEQ * D_INNER;        // L x 96
    float* dtb  = xdbl + (size_t)LSEQ * (DT_RANK + 2 * D_STATE); // L x D_INNER

    dim3 blk(128, 1, 1);   // 4 waves, stacked along M (block tile 128x64)

    // 1) xz = x @ W_in^T   (M=2048, N=4096, K=1024)
    gemm_wmma_f32<<<dim3(4096 / 64, LSEQ / 128), blk, 0, stream>>>(
        x, W_in, nullptr, xz, LSEQ, 2 * D_INNER, D_MODEL, D_MODEL, 0);

    // 2) depthwise causal conv + silu  -> xc
    conv_silu_kernel<<<(LSEQ * D_INNER) / 256, 256, 0, stream>>>(
        xz, conv_w, conv_b, xc);

    // 3) x_dbl = xc @ W_x^T   (N=96, K=2048)
    gemm_wmma_f32<<<dim3((96 + 63) / 64, LSEQ / 128), blk, 0, stream>>>(
        xc, W_x, nullptr, xdbl, LSEQ, DT_RANK + 2 * D_STATE, D_INNER, D_INNER, 0);

    // 4) dt = softplus(dt_low @ W_dt^T + b_dt)   (A = xdbl with lda=96, K=64)
    gemm_wmma_f32<<<dim3(D_INNER / 64, LSEQ / 128), blk, 0, stream>>>(
        xdbl, W_dt, b_dt, dtb, LSEQ, D_INNER, DT_RANK, DT_RANK + 2 * D_STATE, 1);

    // 5) selective scan + D skip + silu(z) gate, y written over xc
    ssm_scan_kernel<<<D_INNER / 256, 256, 0, stream>>>(
        xdbl, dtb, xz, A_log, D_param, xc);

    // 6) out = y @ W_out^T   (N=1024, K=2048)
    gemm_wmma_f32<<<dim3(D_MODEL / 64, LSEQ / 128), blk, 0, stream>>>(
        xc, W_out, nullptr, out, LSEQ, D_MODEL, D_INNER, D_INNER, 0);
}